// CantileverPINN_vanilla_HPT_29437705847120
// MI455X (gfx1250) — compile-verified
//
#include <hip/hip_runtime.h>
#include <hip/hip_bf16.h>

typedef __attribute__((ext_vector_type(2))) float v2f;
typedef __attribute__((ext_vector_type(8))) float v8f;

// D(16x16,f32) = A(16x4,f32) x B(4x16,f32) + C   — CDNA5 V_WMMA_F32_16X16X4_F32
__device__ __forceinline__ v8f wmma4_f32(v2f a, v2f b, v8f c) {
    return __builtin_amdgcn_wmma_f32_16x16x4_f32(
        /*neg_a=*/false, a, /*neg_b=*/false, b,
        /*c_mod=*/(short)0, c, /*reuse_a=*/false, /*reuse_b=*/false);
}

// Hardware transcendental tanh (CDNA5 TRANS32 op). The trailing v_nop satisfies
// the ISA rule: 1 independent op / V_NOP after a TRANS op before its result is used.
__device__ __forceinline__ float htanh(float x) {
    float y;
    asm volatile("v_tanh_f32 %0, %1\n\tv_nop" : "=v"(y) : "v"(x));
    return y;
}

// Degree-4 Taylor jet of tanh: y(t)=tanh(u(t)),  y' = (1-y^2) u'
__device__ __forceinline__ void tanh_jet(float u0, float u1, float u2, float u3, float u4,
                                         float& y0, float& y1, float& y2, float& y3, float& y4) {
    y0 = htanh(u0);
    float v0 = 1.f - y0 * y0;
    y1 = u1 * v0;
    float v1 = -2.f * y0 * y1;
    y2 = 0.5f * u1 * v1 + u2 * v0;
    float v2 = -(2.f * y0 * y2 + y1 * y1);
    y3 = (u1 * v2 + 2.f * u2 * v1 + 3.f * u3 * v0) * (1.f / 3.f);
    float v3 = -(2.f * y0 * y3 + 2.f * y1 * y2);
    y4 = 0.25f * (u1 * v3 + 2.f * u2 * v2 + 3.f * u3 * v1 + 4.f * u4 * v0);
}

#define WPB 4  // waves per block (wave32)

// d_ws float offsets. ws[0..7] = accumulator + bc scalars. All v8f slots 32B-aligned.
#define TAB_W1 16    // 16 (padded)
#define TAB_B1 32    // 16
#define TAB_B2 48    // 32
#define TAB_B3 80    // 64
#define TAB_W4 144   // 64
#define TAB_A2 208   // 512  floats: W2^T A-operands, v2f [u][lane][kc], u<2, kc<4
#define TAB_A3 720   // 2048 floats: W3^T A-operands, v2f [u][lane][kc], u<4, kc<8

// Build zero-padded, pre-swizzled (A-operand per-lane layout) weight tables.
__global__ void prep_kernel(const float* __restrict__ W1, const float* __restrict__ b1,
                            const float* __restrict__ W2, const float* __restrict__ b2,
                            const float* __restrict__ W3, const float* __restrict__ b3,
                            const float* __restrict__ W4, float* __restrict__ ws) {
    int t = threadIdx.x;  // 256 threads, 1 block
    if (t < 8) ws[t] = 0.f;
    if (t < 16) {
        ws[TAB_W1 + t] = (t < 15) ? W1[t] : 0.f;
        ws[TAB_B1 + t] = (t < 15) ? b1[t] : 0.f;
    }
    if (t < 32) ws[TAB_B2 + t] = (t < 30) ? b2[t] : 0.f;
    if (t < 64) {
        ws[TAB_B3 + t] = (t < 60) ? b3[t] : 0.f;
        ws[TAB_W4 + t] = (t < 60) ? W4[t] : 0.f;
    }
    for (int i = t; i < 512; i += 256) {        // layer-2 A operands
        int comp = i & 1, p = i >> 1;
        int kc = p & 3, lane = (p >> 2) & 31, u = p >> 7;
        int k = 4 * kc + 2 * (lane >> 4) + comp;
        int m = 16 * u + (lane & 15);
        ws[TAB_A2 + i] = (k < 15 && m < 30) ? W2[k * 30 + m] : 0.f;
    }
    for (int i = t; i < 2048; i += 256) {       // layer-3 A operands
        int comp = i & 1, p = i >> 1;
        int kc = p & 7, lane = (p >> 3) & 31, u = p >> 8;
        int k = 4 * kc + 2 * (lane >> 4) + comp;
        int m = 16 * u + (lane & 15);
        ws[TAB_A3 + i] = (k < 30 && m < 60) ? W3[k * 60 + m] : 0.f;
    }
}

// Per-point 4th derivative of the MLP via WMMA-propagated jets; reduce sum((24*c4+1)^2).
__global__ __launch_bounds__(128) void pde_kernel(
    const float* __restrict__ x, float* __restrict__ ws, int n)
{
    // Per-wave staging: [wave][coeff][32 rows x 16 points]
    __shared__ float lds[WPB][5][512];

    const int tid  = threadIdx.x;
    const int lane = tid & 31;
    const int wave = tid >> 5;
    const int half = lane >> 4;   // 0: lanes 0-15, 1: lanes 16-31
    const int lm   = lane & 15;   // point column within tile
    const int base = (blockIdx.x * WPB + wave) * 16;
    if (base >= n) return;

    const v2f* __restrict__ A2t = (const v2f*)(ws + TAB_A2);
    const v2f* __restrict__ A3t = (const v2f*)(ws + TAB_A3);

    __builtin_prefetch(ws + TAB_A3, 0, 1);

    const float xv = x[base + lm];

    // ---------- layer 1 (1 -> 16 padded), jets in C-layout (M=unit, N=point) ----------
    v8f w1t = *(const v8f*)(ws + TAB_W1 + 8 * half);
    v8f b1t = *(const v8f*)(ws + TAB_B1 + 8 * half);
    float h1[5][8];
#pragma unroll
    for (int v = 0; v < 8; ++v) {
        float u1 = w1t[v];
        float u0 = xv * u1 + b1t[v];
        tanh_jet(u0, u1, 0.f, 0.f, 0.f,
                 h1[0][v], h1[1][v], h1[2][v], h1[3][v], h1[4][v]);
    }
    // stage h1^T rows 0..15 (row = unit, col = point)
#pragma unroll
    for (int j = 0; j < 5; ++j)
#pragma unroll
        for (int v = 0; v < 8; ++v)
            lds[wave][j][(v + 8 * half) * 16 + lm] = h1[j][v];
    asm volatile("s_wait_dscnt 0" ::: "memory");

    // Hoist all layer-2 B operands (depend only on j,kc — shared across u tiles)
    v2f B2[5][4];
#pragma unroll
    for (int j = 0; j < 5; ++j)
#pragma unroll
        for (int kc = 0; kc < 4; ++kc) {
            int k0 = 4 * kc + 2 * half;
            B2[j][kc].x = lds[wave][j][k0 * 16 + lm];
            B2[j][kc].y = lds[wave][j][(k0 + 1) * 16 + lm];
        }

    // ---------- layer 2 (16 -> 32 padded): D[u] = W2^T x H1^T per Taylor coefficient ----------
    float h2[5][2][8];
#pragma unroll
    for (int u = 0; u < 2; ++u) {
        v8f d[5];
        d[0] = *(const v8f*)(ws + TAB_B2 + 16 * u + 8 * half);
#pragma unroll
        for (int j = 1; j < 5; ++j)
#pragma unroll
            for (int v = 0; v < 8; ++v) d[j][v] = 0.f;
#pragma unroll
        for (int kc = 0; kc < 4; ++kc) {
            v2f a = A2t[(u * 32 + lane) * 4 + kc];
#pragma unroll
            for (int j = 0; j < 5; ++j) d[j] = wmma4_f32(a, B2[j][kc], d[j]);
        }
#pragma unroll
        for (int v = 0; v < 8; ++v)
            tanh_jet(d[0][v], d[1][v], d[2][v], d[3][v], d[4][v],
                     h2[0][u][v], h2[1][u][v], h2[2][u][v], h2[3][u][v], h2[4][u][v]);
    }

    // stage h2^T rows 0..31
    asm volatile("s_wait_dscnt 0" ::: "memory");
#pragma unroll
    for (int j = 0; j < 5; ++j)
#pragma unroll
        for (int u = 0; u < 2; ++u)
#pragma unroll
            for (int v = 0; v < 8; ++v)
                lds[wave][j][(16 * u + v + 8 * half) * 16 + lm] = h2[j][u][v];
    asm volatile("s_wait_dscnt 0" ::: "memory");

    // Hoist all layer-3 B operands (shared across the 4 u tiles: 40 pairs instead of 160)
    v2f B3[5][8];
#pragma unroll
    for (int j = 0; j < 5; ++j)
#pragma unroll
        for (int kc = 0; kc < 8; ++kc) {
            int k0 = 4 * kc + 2 * half;
            B3[j][kc].x = lds[wave][j][k0 * 16 + lm];
            B3[j][kc].y = lds[wave][j][(k0 + 1) * 16 + lm];
        }

    // ---------- layer 3 (32 -> 64 padded) + layer 4 (64 -> 1, c4 only) ----------
    float partial4 = 0.f;
#pragma unroll
    for (int u = 0; u < 4; ++u) {
        v8f d[5];
        d[0] = *(const v8f*)(ws + TAB_B3 + 16 * u + 8 * half);
#pragma unroll
        for (int j = 1; j < 5; ++j)
#pragma unroll
            for (int v = 0; v < 8; ++v) d[j][v] = 0.f;
#pragma unroll
        for (int kc = 0; kc < 8; ++kc) {
            v2f a = A3t[(u * 32 + lane) * 8 + kc];
#pragma unroll
            for (int j = 0; j < 5; ++j) d[j] = wmma4_f32(a, B3[j][kc], d[j]);
        }
        v8f w4t = *(const v8f*)(ws + TAB_W4 + 16 * u + 8 * half);
#pragma unroll
        for (int v = 0; v < 8; ++v) {
            float y0, y1, y2, y3, y4;
            tanh_jet(d[0][v], d[1][v], d[2][v], d[3][v], d[4][v], y0, y1, y2, y3, y4);
            partial4 += y4 * w4t[v];
        }
    }
    // lanes l and l+16 hold complementary unit subsets of the same point
    float c4 = partial4 + __shfl_xor(partial4, 16, 32);
    float resid = 24.f * c4 + 1.0f;            // w'''' + P/(E*I)
    float sq = resid * resid;                  // duplicated across lane halves
#pragma unroll
    for (int off = 16; off > 0; off >>= 1) sq += __shfl_xor(sq, off, 32);
    if (lane == 0) atomicAdd(&ws[0], 0.5f * sq);  // halve: each point counted twice
}

// Scalar jets at x=0 (thread 0) and x=L (thread 1): constants for every output row.
__global__ void bc_kernel(const float* __restrict__ W1, const float* __restrict__ b1,
                          const float* __restrict__ W2, const float* __restrict__ b2,
                          const float* __restrict__ W3, const float* __restrict__ b3,
                          const float* __restrict__ W4, const float* __restrict__ b4,
                          float* __restrict__ ws)
{
    int t = blockIdx.x * blockDim.x + threadIdx.x;
    if (t >= 2) return;
    float xv = (t == 0) ? 0.f : 1.0f;  // LENGTH = 1
    float h[5][64], p[5][64];

    for (int m = 0; m < 15; ++m) {
        float u0 = xv * W1[m] + b1[m], u1 = W1[m];
        tanh_jet(u0, u1, 0.f, 0.f, 0.f, h[0][m], h[1][m], h[2][m], h[3][m], h[4][m]);
    }
    for (int m = 0; m < 30; ++m) {
        float s[5];
        for (int j = 0; j < 5; ++j) s[j] = (j == 0) ? b2[m] : 0.f;
        for (int k = 0; k < 15; ++k) {
            float w = W2[k * 30 + m];
            for (int j = 0; j < 5; ++j) s[j] += h[j][k] * w;
        }
        tanh_jet(s[0], s[1], s[2], s[3], s[4], p[0][m], p[1][m], p[2][m], p[3][m], p[4][m]);
    }
    for (int m = 0; m < 60; ++m) {
        float s[5];
        for (int j = 0; j < 5; ++j) s[j] = (j == 0) ? b3[m] : 0.f;
        for (int k = 0; k < 30; ++k) {
            float w = W3[k * 60 + m];
            for (int j = 0; j < 5; ++j) s[j] += p[j][k] * w;
        }
        tanh_jet(s[0], s[1], s[2], s[3], s[4], h[0][m], h[1][m], h[2][m], h[3][m], h[4][m]);
    }
    float c[5];
    for (int j = 0; j < 5; ++j) {
        c[j] = (j == 0) ? b4[0] : 0.f;
        for (int k = 0; k < 60; ++k) c[j] += h[j][k] * W4[k];
    }
    if (t == 0) { ws[1] = c[0]; ws[2] = c[1]; }             // w(0), w'(0)
    else        { ws[3] = 2.f * c[2]; ws[4] = 6.f * c[3]; } // M(L)=w''(L), V(L)=w'''(L)
}

// All output arrays are constant fills (loss/bcs broadcast); pde_loss is a scalar.
__global__ void fill_kernel(const float* __restrict__ ws, float* __restrict__ out, int n) {
    float pde = ws[0] / (float)n;
    float l0 = ws[1] * ws[1];
    float l1 = ws[2] * ws[2];
    float l2 = ws[3] * ws[3];
    float l3 = ws[4] * ws[4];
    float lossv = pde + l0 + l1 + l2 + l3;
    int i = blockIdx.x * blockDim.x + threadIdx.x;
    if (i < n) {
        out[i]             = lossv;
        out[n + 1 + i]     = l0;
        out[2 * n + 1 + i] = l1;
        out[3 * n + 1 + i] = l2;
        out[4 * n + 1 + i] = l3;
    }
    if (i == 0) out[n] = pde;
}

extern "C" void kernel_launch(void* const* d_in, const int* in_sizes, int n_in,
                              void* d_out, int out_size, void* d_ws, size_t ws_size,
                              hipStream_t stream) {
    const float* x  = (const float*)d_in[0];
    const float* W1 = (const float*)d_in[1];
    const float* b1 = (const float*)d_in[2];
    const float* W2 = (const float*)d_in[3];
    const float* b2 = (const float*)d_in[4];
    const float* W3 = (const float*)d_in[5];
    const float* b3 = (const float*)d_in[6];
    const float* W4 = (const float*)d_in[7];
    const float* b4 = (const float*)d_in[8];
    float* out = (float*)d_out;
    float* ws  = (float*)d_ws;
    const int n = in_sizes[0];

    prep_kernel<<<1, 256, 0, stream>>>(W1, b1, W2, b2, W3, b3, W4, ws);
    const int blocks = (n + (WPB * 16) - 1) / (WPB * 16);
    pde_kernel<<<blocks, WPB * 32, 0, stream>>>(x, ws, n);
    bc_kernel<<<1, 64, 0, stream>>>(W1, b1, W2, b2, W3, b3, W4, b4, ws);
    fill_kernel<<<(n + 255) / 256, 256, 0, stream>>>(ws, out, n);
}